// PerceptualCost2_75694503625122
// MI455X (gfx1250) — compile-verified
//
#include <hip/hip_runtime.h>
#include <hip/hip_bf16.h>
#include <math.h>

#define EPSV 1e-10f
#define IMG_ELEMS (512 * 512)

typedef __attribute__((ext_vector_type(2))) float v2f;
typedef __attribute__((ext_vector_type(8))) float v8f;
typedef int v4i __attribute__((vector_size(16)));   // matches builtin param type

#define AS1 __attribute__((address_space(1)))
#define AS3 __attribute__((address_space(3)))

#if defined(__has_builtin)
#if __has_builtin(__builtin_amdgcn_global_load_async_to_lds_b128)
#define HAVE_ASYNC_LDS 1
#endif
#if __has_builtin(__builtin_amdgcn_s_wait_asynccnt)
#define HAVE_WAIT_ASYNC 1
#endif
#endif
#ifndef HAVE_ASYNC_LDS
#define HAVE_ASYNC_LDS 0
#endif
#ifndef HAVE_WAIT_ASYNC
#define HAVE_WAIT_ASYNC 0
#endif

static __device__ __forceinline__ void wait_async0() {
#if HAVE_WAIT_ASYNC
    __builtin_amdgcn_s_wait_asynccnt(0);
#else
    asm volatile("s_wait_asynccnt 0x0" ::: "memory");
#endif
}

#if HAVE_ASYNC_LDS
// Direct memory -> LDS async copy (tracked by ASYNCcnt), bypasses VGPRs.
static __device__ __forceinline__ void async_b128(const float* g, float* l) {
    __builtin_amdgcn_global_load_async_to_lds_b128(
        (AS1 v4i*)(uintptr_t)g,
        (AS3 v4i*)(uint32_t)(uintptr_t)l, /*offset=*/0, /*cpol=*/0);
}
#endif

// D = A(16x4) * B(4x16) + C, fp32 WMMA (CDNA5 V_WMMA_F32_16X16X4_F32)
static __device__ __forceinline__ v8f wmma4(v2f a, v2f b, v8f c) {
    return __builtin_amdgcn_wmma_f32_16x16x4_f32(
        /*neg_a=*/false, a, /*neg_b=*/false, b,
        /*c_mod=*/(short)0, c, /*reuse_a=*/false, /*reuse_b=*/false);
}

// blockdiag(F8,F8)[r][c], F8[j][k] = exp(-2*pi*i*j*k/8)
static __device__ __forceinline__ void f8_entry(int r, int c, float& re, float& im) {
    if ((r >> 3) != (c >> 3)) { re = 0.f; im = 0.f; return; }
    int idx = ((r & 7) * (c & 7)) & 7;
    float ang = -0.78539816339744831f * (float)idx;  // -pi/4 * idx
    re = __cosf(ang);
    im = __sinf(ang);
}

// blockdiag(R,R)[k][n], R = 8x8, R[k][n] = exp(-2*pi*i*k*n/8)/64 for n<5, else 0 (pad)
static __device__ __forceinline__ void r8_entry(int k, int n, float& re, float& im) {
    int nb = n & 7;
    if (((k >> 3) != (n >> 3)) || (nb >= 5)) { re = 0.f; im = 0.f; return; }
    int idx = ((k & 7) * nb) & 7;
    float ang = -0.78539816339744831f * (float)idx;
    re = __cosf(ang) * (1.0f / 64.0f);
    im = __sinf(ang) * (1.0f / 64.0f);
}

__global__ void ws_init_kernel(float* ws, int n_floats) {
    int i = blockIdx.x * blockDim.x + threadIdx.x;
    if (i < n_floats) ws[i] = 0.0f;
}

// Phase 1: sum of DC amplitudes of the target -> ws[0]
__global__ void __launch_bounds__(256) dc_sum_kernel(const float* __restrict__ tgt, float* ws) {
    int b = blockIdx.x * blockDim.x + threadIdx.x;   // global block index over N*4096
    int n = b >> 12;
    int k = b & 4095;
    int by = k >> 6, bx = k & 63;
    const float* p = tgt + ((size_t)n << 18) + (size_t)(by * 8) * 512 + bx * 8;
    float s = 0.0f;
#pragma unroll
    for (int r = 0; r < 8; ++r) {
        float4 v0 = *(const float4*)(p + (size_t)r * 512);
        float4 v1 = *(const float4*)(p + (size_t)r * 512 + 4);
        s += v0.x + v0.y + v0.z + v0.w + v1.x + v1.y + v1.z + v1.w;
    }
    float dc = s * (1.0f / 64.0f);
    float amp = sqrtf((dc + EPSV) * (dc + EPSV) + EPSV * EPSV);
#pragma unroll
    for (int off = 16; off > 0; off >>= 1) amp += __shfl_xor(amp, off, 32);
    __shared__ float red[8];
    int lane = threadIdx.x & 31, wid = threadIdx.x >> 5;
    if (lane == 0) red[wid] = amp;
    __syncthreads();
    if (threadIdx.x == 0) {
        float t = 0.0f;
#pragma unroll
        for (int i = 0; i < 8; ++i) t += red[i];
        atomicAdd(&ws[0], t);
    }
}

// Phase 2: one wave32 per 16x16 tile (= 2x2 group of 8x8 blocks) of one image pair.
// Tiles staged into LDS with GLOBAL_LOAD_ASYNC_TO_LDS_B128 (ASYNCcnt), block-FFT via
// fp32 WMMA chains with block-diagonal DFT matrices, then pointwise Watson terms.
__global__ void __launch_bounds__(256) watson_main_kernel(
    const float* __restrict__ inp, const float* __restrict__ tgt,
    const float* __restrict__ t_tild, const float* __restrict__ alpha_p,
    const float* __restrict__ w_tild_p, const float* __restrict__ beta_p,
    const float* __restrict__ wp_tild, float* ws, int nimg, float inv_count) {
    __shared__ float sh_x[8][256];   // per-wave 16x16 staging tile
    __shared__ float sh_y[8][512];   // per-wave Yr (0..255) / Yi (256..511)
    __shared__ float sh_t[40];       // exp(t_tild)
    __shared__ float sh_wp[40];      // exp(w_phase_tild)

    if (threadIdx.x < 40) {
        sh_t[threadIdx.x]  = __expf(t_tild[threadIdx.x]);
        sh_wp[threadIdx.x] = __expf(wp_tild[threadIdx.x]);
    }
    __syncthreads();

    int lane = threadIdx.x & 31;
    int wid  = threadIdx.x >> 5;
    int tile = blockIdx.x * 8 + wid;         // N*1024 tiles total
    int n  = tile >> 10;
    int ti = tile & 1023;
    int tby = ti >> 5, tbx = ti & 31;

    int lh = lane >> 4;       // lane half (0/1)
    int ln = lane & 15;       // lane-in-half

    // Constant DFT A-fragments for the column FFT: chunk q, K = 2*lh + {0,1}
    v2f aFr[4], aFi[4];
#pragma unroll
    for (int q = 0; q < 4; ++q) {
        int c0 = 4 * q + 2 * lh;
        float r0, i0, r1, i1;
        f8_entry(ln, c0, r0, i0);
        f8_entry(ln, c0 + 1, r1, i1);
        aFr[q].x = r0; aFr[q].y = r1;
        aFi[q].x = i0; aFi[q].y = i1;
    }
    // Constant rFFT B-fragments (and negated imag) for the row transform
    v2f bRr[4], bRi[4], bRiN[4];
#pragma unroll
    for (int q = 0; q < 4; ++q) {
        int k0 = 4 * q + 2 * lh;
        float r0, i0, r1, i1;
        r8_entry(k0, ln, r0, i0);
        r8_entry(k0 + 1, ln, r1, i1);
        bRr[q].x = r0;  bRr[q].y = r1;
        bRi[q].x = i0;  bRi[q].y = i1;
        bRiN[q].x = -i0; bRiN[q].y = -i1;
    }

    v8f z0r = {}, z0i = {}, z1r = {}, z1i = {};

#pragma unroll
    for (int img = 0; img < 2; ++img) {
        const float* src = (img == 0) ? tgt : inp;
        const float* base = src + ((size_t)n << 18) + (size_t)(tby * 16) * 512 + tbx * 16;
        // Stage 16x16 tile into LDS (rows are 64B-contiguous in memory)
#pragma unroll
        for (int rep = 0; rep < 2; ++rep) {
            int r = rep * 8 + (lane >> 2);
            int q = lane & 3;
            const float* gp = base + (size_t)r * 512 + q * 4;
            float* lp = &sh_x[wid][r * 16 + q * 4];
#if HAVE_ASYNC_LDS
            async_b128(gp, lp);                       // global -> LDS, ASYNCcnt
#else
            float4 v = *(const float4*)gp;
            *(float4*)lp = v;
#endif
        }
#if HAVE_ASYNC_LDS
        wait_async0();
#endif
        __syncthreads();

        // Column FFT: Y = blockdiag(F8,F8) * X   (8 WMMAs)
        v8f yr = {}, yi = {};
#pragma unroll
        for (int q = 0; q < 4; ++q) {
            int k0 = 4 * q + 2 * lh;
            v2f b;
            b.x = sh_x[wid][k0 * 16 + ln];
            b.y = sh_x[wid][(k0 + 1) * 16 + ln];
            yr = wmma4(aFr[q], b, yr);
            yi = wmma4(aFi[q], b, yi);
        }
        // Spill Y (accumulator layout: row = i + 8*lh, col = ln)
#pragma unroll
        for (int i = 0; i < 8; ++i) {
            sh_y[wid][(i + 8 * lh) * 16 + ln]       = yr[i];
            sh_y[wid][256 + (i + 8 * lh) * 16 + ln] = yi[i];
        }
        __syncthreads();

        // Row rFFT: Z = Y * blockdiag(R,R)/64, complex (16 WMMAs)
        v8f zr = {}, zi = {};
#pragma unroll
        for (int q = 0; q < 4; ++q) {
            int k0 = 4 * q + 2 * lh;
            v2f aYr, aYi;
            aYr.x = sh_y[wid][ln * 16 + k0];
            aYr.y = sh_y[wid][ln * 16 + k0 + 1];
            aYi.x = sh_y[wid][256 + ln * 16 + k0];
            aYi.y = sh_y[wid][256 + ln * 16 + k0 + 1];
            zr = wmma4(aYr, bRr[q], zr);
            zr = wmma4(aYi, bRiN[q], zr);
            zi = wmma4(aYr, bRi[q], zi);
            zi = wmma4(aYi, bRr[q], zi);
        }
        if (img == 0) { z0r = zr; z0i = zi; } else { z1r = zr; z1i = zi; }
        __syncthreads();   // before next image reuses sh_x / sh_y
    }

    // ---- pointwise Watson math ----
    float alpha = alpha_p[0];
    float beta  = beta_p[0];
    float w     = 1.0f / (1.0f + __expf(-w_tild_p[0]));
    float avgl  = ws[0] * inv_count;

    // DC amplitude of the target block owning this lane (vgpr0 of lane & 24)
    float a00 = sqrtf((z0r[0] + EPSV) * (z0r[0] + EPSV) + (z0i[0] + EPSV) * (z0i[0] + EPSV));
    float dc  = __shfl(a00, lane & 24, 32);
    float lum = __powf((dc + EPSV) / (avgl + EPSV), alpha);

    int kc = ln & 7;
    bool valid = (kc < 5);
    int kcc = valid ? kc : 4;

    float wsum = 0.0f, psum = 0.0f;
#pragma unroll
    for (int i = 0; i < 8; ++i) {
        float c0re = z0r[i], c0im = z0i[i];
        float c1re = z1r[i], c1im = z1i[i];
        float a0 = sqrtf((c0re + EPSV) * (c0re + EPSV) + (c0im + EPSV) * (c0im + EPSV));
        float a1 = sqrtf((c1re + EPSV) * (c1re + EPSV) + (c1im + EPSV) * (c1im + EPSV));
        float t_l = sh_t[i * 5 + kcc] * lum;
        float tmp = __powf(a0 + EPSV, w) * __powf(t_l, 1.0f - w);
        float d   = tmp - t_l;
        float s   = t_l + d * (1.0f / (1.0f + __expf(-d)));   // softmax(2) blend
        float wt  = __powf(fabsf((a0 - a1) / s) + EPSV, beta) + EPSV;
        float p0  = atan2f(c0im, c0re + EPSV);
        float p1  = atan2f(c1im, c1re + EPSV);
        float pd  = acosf(__cosf(p0 - p1) * (1.0f - 1e-7f)) * sh_wp[i * 5 + kcc];
        if (valid) { wsum += wt; psum += pd; }
    }
#pragma unroll
    for (int off = 16; off > 0; off >>= 1) {
        wsum += __shfl_xor(wsum, off, 32);
        psum += __shfl_xor(psum, off, 32);
    }
    if (lane == 0) {
        atomicAdd(&ws[1 + n], wsum);
        atomicAdd(&ws[1 + nimg + n], psum);
    }
}

__global__ void finalize_kernel(const float* ws, const float* beta_p, float* out, int nimg) {
    int n = blockIdx.x * blockDim.x + threadIdx.x;
    if (n < nimg) {
        float beta = beta_p[0];
        out[n] = __powf(ws[1 + n], 1.0f / beta) + ws[1 + nimg + n];
    }
}

extern "C" void kernel_launch(void* const* d_in, const int* in_sizes, int n_in,
                              void* d_out, int out_size, void* d_ws, size_t ws_size,
                              hipStream_t stream) {
    const float* inp     = (const float*)d_in[0];
    const float* tgt     = (const float*)d_in[1];
    const float* t_tild  = (const float*)d_in[2];
    const float* alpha   = (const float*)d_in[3];
    const float* w_tild  = (const float*)d_in[4];
    const float* beta    = (const float*)d_in[5];
    const float* wp_tild = (const float*)d_in[6];
    float* out = (float*)d_out;
    float* ws  = (float*)d_ws;

    int nimg = in_sizes[0] / IMG_ELEMS;          // N images (512x512 each)
    int nblocks_total = nimg * 4096;             // 8x8 blocks over all images
    float inv_count = 1.0f / (float)nblocks_total;
    int ws_floats = 1 + 2 * nimg;

    ws_init_kernel<<<(ws_floats + 255) / 256, 256, 0, stream>>>(ws, ws_floats);
    dc_sum_kernel<<<nblocks_total / 256, 256, 0, stream>>>(tgt, ws);
    // N*1024 tiles, 8 waves (tiles) per 256-thread workgroup
    watson_main_kernel<<<nimg * 128, 256, 0, stream>>>(
        inp, tgt, t_tild, alpha, w_tild, beta, wp_tild, ws, nimg, inv_count);
    finalize_kernel<<<(nimg + 63) / 64, 64, 0, stream>>>(ws, beta, out, nimg);
}